// RNN_ENCODER_8100308320649
// MI455X (gfx1250) — compile-verified
//
#include <hip/hip_runtime.h>
#include <cmath>

typedef __attribute__((ext_vector_type(2))) float v2f;
typedef __attribute__((ext_vector_type(8))) float v8f;

#define B_   4096
#define T_   18
#define NIN  300
#define NH   64
#define NG   256   // 4*NH, gate width

__device__ __forceinline__ float sigmoidf_(float x) {
    return 1.0f / (1.0f + __expf(-x));
}

// -----------------------------------------------------------------------------
// Kernel 1: gx[dir][t][b][0:256] = E[captions[b][t]] @ W_dir + bias_dir
// Grid: (4608, 2) = (T*B/16 row-tiles, dir). Block: 256 threads = 8 waves.
// Each block computes a 16 x 256 output tile with v_wmma_f32_16x16x4_f32.
// -----------------------------------------------------------------------------
__global__ __launch_bounds__(256) void gemm_xw(
    const int*   __restrict__ captions,  // [B][T]
    const float* __restrict__ E,         // [NTOKEN][300]
    const float* __restrict__ W_f,       // [300][256]
    const float* __restrict__ b_f,       // [256]
    const float* __restrict__ W_b,
    const float* __restrict__ b_b,
    float*       __restrict__ gx)        // [2][T][B][256]
{
    __shared__ float sA[16 * 304];       // gathered embedding rows, padded stride
    __shared__ int   sTok[16];

    const int tid = threadIdx.x;
    const int dir = blockIdx.y;
    const int t   = blockIdx.x / (B_ / 16);
    const int b0  = (blockIdx.x % (B_ / 16)) * 16;

    const float* W    = dir ? W_b : W_f;
    const float* bias = dir ? b_b : b_f;

    if (tid < 16) sTok[tid] = captions[(b0 + tid) * T_ + t];
    __syncthreads();

    // Gather 16 embedding rows (16 threads per row, coalesced 64B runs)
    {
        const int r = tid >> 4;
        const int l = tid & 15;
        const float* erow = E + (size_t)sTok[r] * NIN;
        for (int k = l; k < NIN; k += 16) sA[r * 304 + k] = erow[k];
    }
    __syncthreads();

    const int wave = tid >> 5;      // 0..7
    const int lane = tid & 31;
    const int half = lane >> 4;     // 0 or 1
    const int l16  = lane & 15;
    const int n0   = wave * 32;     // this wave owns cols [n0, n0+32)

    v8f acc0 = {0.f,0.f,0.f,0.f,0.f,0.f,0.f,0.f};
    v8f acc1 = {0.f,0.f,0.f,0.f,0.f,0.f,0.f,0.f};

    for (int kk = 0; kk < NIN / 4; ++kk) {
        const int k0 = kk * 4 + 2 * half;
        v2f a;
        a.x = sA[l16 * 304 + k0];
        a.y = sA[l16 * 304 + k0 + 1];
        const float* wr = W + (size_t)k0 * NG;
        v2f bA, bB;
        bA.x = wr[n0 + l16];
        bA.y = wr[NG + n0 + l16];
        bB.x = wr[n0 + 16 + l16];
        bB.y = wr[NG + n0 + 16 + l16];
        acc0 = __builtin_amdgcn_wmma_f32_16x16x4_f32(false, a, false, bA,
                                                     (short)0, acc0, false, false);
        acc1 = __builtin_amdgcn_wmma_f32_16x16x4_f32(false, a, false, bB,
                                                     (short)0, acc1, false, false);
    }

    float* out = gx + (((size_t)dir * T_ + t) * B_ + b0) * NG;
    const float bv0 = bias[n0 + l16];
    const float bv1 = bias[n0 + 16 + l16];
#pragma unroll
    for (int i = 0; i < 8; ++i) {
        const int m = i + 8 * half;          // output row within tile
        out[(size_t)m * NG + n0 + l16]      = acc0[i] + bv0;
        out[(size_t)m * NG + n0 + 16 + l16] = acc1[i] + bv1;
    }
}

// -----------------------------------------------------------------------------
// Kernel 2: per-direction LSTM scan. Each block owns 16 batch rows; h lives in
// LDS, c in registers; per step h@U is a 16x256x64 WMMA GEMM (U streamed from
// L2: each element used exactly once per step, so LDS staging buys nothing).
// Grid: (256, 2). Block: 256 threads = 8 waves. No inter-block sync needed.
// -----------------------------------------------------------------------------
__global__ __launch_bounds__(256) void lstm_scan(
    const float* __restrict__ h0_f, const float* __restrict__ c0_f,
    const float* __restrict__ h0_b, const float* __restrict__ c0_b,
    const float* __restrict__ U_f,  const float* __restrict__ U_b,  // [64][256]
    const float* __restrict__ gx,   // [2][T][B][256]
    float*       __restrict__ words,  // [B][128][T]
    float*       __restrict__ sent)   // [B][128]
{
    __shared__ float hs[16 * 65];     // h state, padded stride (bank-conflict free)
    __shared__ float zb[16 * 256];    // h@U tile buffer

    const int tid = threadIdx.x;
    const int dir = blockIdx.y;
    const int r0  = blockIdx.x * 16;  // batch rows r0..r0+15

    const float* U  = dir ? U_b  : U_f;
    const float* h0 = dir ? h0_b : h0_f;
    const float* c0 = dir ? c0_b : c0_f;
    const float* gxd = gx + (size_t)dir * T_ * B_ * NG;

    // Stage initial h, c
    for (int idx = tid; idx < 16 * NH; idx += 256) {
        const int r = idx >> 6, j = idx & 63;
        hs[r * 65 + j] = h0[(size_t)(r0 + r) * NH + j];
    }
    float creg[4];
#pragma unroll
    for (int q = 0; q < 4; ++q) {
        const int idx = tid + 256 * q;
        const int r = idx >> 6, j = idx & 63;
        creg[q] = c0[(size_t)(r0 + r) * NH + j];
    }

    const int wave = tid >> 5, lane = tid & 31;
    const int half = lane >> 4, l16 = lane & 15;
    const int n0   = wave * 32;
    const int j    = tid & 63;   // elementwise hidden index
    const int rb   = tid >> 6;   // elementwise row base (0..3)

    for (int tt = 0; tt < T_; ++tt) {
        const int t = dir ? (T_ - 1 - tt) : tt;
        __syncthreads();   // hs from previous step visible; zb free to overwrite

        v8f acc0 = {0.f,0.f,0.f,0.f,0.f,0.f,0.f,0.f};
        v8f acc1 = {0.f,0.f,0.f,0.f,0.f,0.f,0.f,0.f};
#pragma unroll
        for (int kk = 0; kk < NH / 4; ++kk) {
            const int k0 = kk * 4 + 2 * half;
            v2f a;
            a.x = hs[l16 * 65 + k0];
            a.y = hs[l16 * 65 + k0 + 1];
            const float* ur = U + (size_t)k0 * NG;
            v2f bA, bB;
            bA.x = ur[n0 + l16];
            bA.y = ur[NG + n0 + l16];
            bB.x = ur[n0 + 16 + l16];
            bB.y = ur[NG + n0 + 16 + l16];
            acc0 = __builtin_amdgcn_wmma_f32_16x16x4_f32(false, a, false, bA,
                                                         (short)0, acc0, false, false);
            acc1 = __builtin_amdgcn_wmma_f32_16x16x4_f32(false, a, false, bB,
                                                         (short)0, acc1, false, false);
        }
#pragma unroll
        for (int i = 0; i < 8; ++i) {
            const int m = i + 8 * half;
            zb[m * 256 + n0 + l16]      = acc0[i];
            zb[m * 256 + n0 + 16 + l16] = acc1[i];
        }
        __syncthreads();   // zb complete; hs free to overwrite

        const float* g = gxd + ((size_t)t * B_ + r0) * NG;
#pragma unroll
        for (int q = 0; q < 4; ++q) {
            const int r = rb + 4 * q;
            const float* gr = g + (size_t)r * NG;
            const float zi = zb[r * 256 + j]       + gr[j];
            const float zf = zb[r * 256 + 64 + j]  + gr[64 + j];
            const float zg = zb[r * 256 + 128 + j] + gr[128 + j];
            const float zo = zb[r * 256 + 192 + j] + gr[192 + j];
            const float ig = sigmoidf_(zi);
            const float fg = sigmoidf_(zf);
            const float gg = tanhf(zg);
            const float og = sigmoidf_(zo);
            const float c  = fg * creg[q] + ig * gg;
            creg[q] = c;
            const float h = og * tanhf(c);
            hs[r * 65 + j] = h;
            // words_emb[b][feat][t], feat = dir*64 + j
            words[((size_t)(r0 + r) * 128 + dir * NH + j) * T_ + t] = h;
            if (tt == T_ - 1)
                sent[(size_t)(r0 + r) * 128 + dir * NH + j] = h;
        }
    }
}

extern "C" void kernel_launch(void* const* d_in, const int* in_sizes, int n_in,
                              void* d_out, int out_size, void* d_ws, size_t ws_size,
                              hipStream_t stream) {
    const int*   captions = (const int*)  d_in[0];
    // d_in[1] = cap_lens (unused by reference compute)
    const float* h0_f = (const float*)d_in[2];
    const float* c0_f = (const float*)d_in[3];
    const float* h0_b = (const float*)d_in[4];
    const float* c0_b = (const float*)d_in[5];
    const float* E    = (const float*)d_in[6];
    const float* W_f  = (const float*)d_in[7];
    const float* U_f  = (const float*)d_in[8];
    const float* b_f  = (const float*)d_in[9];
    const float* W_b  = (const float*)d_in[10];
    const float* U_b  = (const float*)d_in[11];
    const float* b_b  = (const float*)d_in[12];

    float* out = (float*)d_out;
    float* gx  = (float*)d_ws;   // [2][T][B][256] f32 = 151 MB scratch

    dim3 blk(256);
    dim3 g1((T_ * B_) / 16, 2);
    gemm_xw<<<g1, blk, 0, stream>>>(captions, E, W_f, b_f, W_b, b_b, gx);

    dim3 g2(B_ / 16, 2);
    lstm_scan<<<g2, blk, 0, stream>>>(h0_f, c0_f, h0_b, c0_b, U_f, U_b, gx,
                                      out, out + (size_t)B_ * 128 * T_);
}